// PlanetoidGCN_15762529976321
// MI455X (gfx1250) — compile-verified
//
#include <hip/hip_runtime.h>

// ---------------------------------------------------------------------------
// GCNConv (no self loops) + bias + PReLU for MI455X (gfx1250, wave32, WMMA).
//   features  : [100000, 512] f32
//   edge_index: [2, 3200000] i64   (row = src, col = dst)
//   W         : [512, 64] f32, b: [64] f32, prelu_a: [1] f32
//   out       : [100000, 64] f32
//
// Phases: zero -> degree -> d^-1/2 -> x = X@W (WMMA f32) -> edge scatter-add
//         (atomics, L2-resident) -> bias+PReLU.
// ---------------------------------------------------------------------------

typedef __attribute__((ext_vector_type(2))) float v2f;
typedef __attribute__((ext_vector_type(8))) float v8f;

#define N_NODES 100000
#define N_EDGES 3200000
#define IN_DIM  512
#define OUT_DIM 64

// ---------------------------------------------------------------------------
// 0) Zero the scatter accumulator (d_out) and the degree array (workspace).
//    float4 stores to cut the grid 4x.
// ---------------------------------------------------------------------------
__global__ __launch_bounds__(256)
void gcn_zero_kernel(float4* __restrict__ out4, float* __restrict__ deg) {
    int i = blockIdx.x * blockDim.x + threadIdx.x;
    if (i < (N_NODES * OUT_DIM) / 4) out4[i] = make_float4(0.f, 0.f, 0.f, 0.f);
    if (i < N_NODES)                 deg[i] = 0.0f;
}

// ---------------------------------------------------------------------------
// 1) In-degree at targets: deg[col] += 1 per edge. Edge stream is read-once
//    (51 MB) -> non-temporal so it does not pollute L2 (keep L2 for x/out).
// ---------------------------------------------------------------------------
__global__ __launch_bounds__(256)
void gcn_degree_kernel(const long long* __restrict__ ei,
                       float* __restrict__ deg) {
    int e = blockIdx.x * blockDim.x + threadIdx.x;
    if (e >= N_EDGES) return;
    long long c = __builtin_nontemporal_load(&ei[(long long)N_EDGES + e]);
    atomicAdd(&deg[(int)c], 1.0f);
}

// ---------------------------------------------------------------------------
// 2) deg -> d^{-1/2} in place (reference: where(deg>0, rsqrt(max(deg,1)), 0)).
// ---------------------------------------------------------------------------
__global__ __launch_bounds__(256)
void gcn_dinv_kernel(float* __restrict__ deg) {
    int i = blockIdx.x * blockDim.x + threadIdx.x;
    if (i >= N_NODES) return;
    float d = deg[i];
    deg[i] = (d > 0.0f) ? rsqrtf(fmaxf(d, 1.0f)) : 0.0f;
}

// ---------------------------------------------------------------------------
// 3) x = X @ W using V_WMMA_F32_16X16X4_F32.
//    Block = 128 threads = 4 waves. Block owns 16 rows (M-tile); wave w owns
//    columns [16w, 16w+16). K=512 looped in steps of 4.
//    A layout (16x4 f32): lane<16 -> K={k0,k0+1}, lane>=16 -> K={k0+2,k0+3},
//    row M = lane%16. B layout (4x16) mirrors A. C/D: M = r + 8*(lane>=16),
//    N = lane%16 in VGPR r of the v8f accumulator.
// ---------------------------------------------------------------------------
__global__ __launch_bounds__(128)
void gcn_gemm_kernel(const float* __restrict__ X,
                     const float* __restrict__ W,
                     float* __restrict__ xout) {
    const int lane = threadIdx.x & 31;
    const int wave = threadIdx.x >> 5;
    const int l15  = lane & 15;
    const int hi   = lane >> 4;            // 0 or 1: selects K-pair within step
    const int m0   = blockIdx.x * 16;      // 100000 = 16 * 6250, no tail
    const int n0   = wave * 16;

    // Per-lane base pointers. A loads are 8B-aligned float2 (k0 and 2*hi even).
    const float* Arow = X + (size_t)(m0 + l15) * IN_DIM + 2 * hi;
    const float* Bcol = W + (size_t)(2 * hi) * OUT_DIM + n0 + l15;

    v8f acc = {};
#pragma unroll 4
    for (int k0 = 0; k0 < IN_DIM; k0 += 4) {
        const float2 av = *reinterpret_cast<const float2*>(Arow + k0);
        v2f a; a.x = av.x; a.y = av.y;                    // (M, k0+2hi), (M, k0+2hi+1)
        v2f b;
        b.x = Bcol[(size_t)k0 * OUT_DIM];                 // (k0+2hi,   N)
        b.y = Bcol[(size_t)(k0 + 1) * OUT_DIM];           // (k0+2hi+1, N)
        acc = __builtin_amdgcn_wmma_f32_16x16x4_f32(
            /*neg_a=*/false, a, /*neg_b=*/false, b,
            /*c_mod=*/(short)0, acc, /*reuse_a=*/false, /*reuse_b=*/false);
    }

    float* dst = xout + (size_t)m0 * OUT_DIM + n0 + l15;
#pragma unroll
    for (int r = 0; r < 8; ++r) {
        dst[(size_t)(r + 8 * hi) * OUT_DIM] = acc[r];     // M = r + 8*hi
    }
}

// ---------------------------------------------------------------------------
// 4) Scatter: out[col, j] += x[row, j] * dinv[row] * dinv[col].
//    64 consecutive threads per edge -> each wave32 maps to exactly one edge
//    (j = 0..31 or 32..63), so `e` is wave-uniform. Force that knowledge with
//    readfirstlane so the edge indices and dinv values become SMEM scalar
//    loads (KMcnt) + SALU math, leaving the vector memory pipe for the
//    perfectly coalesced x gather (128 B/wave) and the f32 atomics.
//    Total: 204.8M atomics, but x (25.6 MB) and out (25.6 MB) both live in
//    the 192 MB L2, so this runs at L2 atomic throughput, not HBM.
// ---------------------------------------------------------------------------
__global__ __launch_bounds__(256)
void gcn_scatter_kernel(const long long* __restrict__ ei,
                        const float* __restrict__ x,
                        const float* __restrict__ dinv,
                        float* __restrict__ out) {
    long long t = (long long)blockIdx.x * blockDim.x + threadIdx.x;
    int e = (int)(t >> 6);
    int j = (int)(t & 63);
    if (e >= N_EDGES) return;
    e = __builtin_amdgcn_readfirstlane(e);        // provably wave-uniform
    int r = (int)ei[e];                           // scalar load (src)
    int c = (int)ei[(long long)N_EDGES + e];      // scalar load (dst)
    float norm = dinv[r] * dinv[c];               // scalar loads + scalar mul
    atomicAdd(&out[(size_t)c * OUT_DIM + j], x[(size_t)r * OUT_DIM + j] * norm);
}

// ---------------------------------------------------------------------------
// 5) out = PReLU(out + b) in place.
// ---------------------------------------------------------------------------
__global__ __launch_bounds__(256)
void gcn_bias_prelu_kernel(float* __restrict__ out,
                           const float* __restrict__ b,
                           const float* __restrict__ prelu_a) {
    int i = blockIdx.x * blockDim.x + threadIdx.x;
    if (i >= N_NODES * OUT_DIM) return;
    float v = out[i] + b[i & (OUT_DIM - 1)];
    out[i] = (v >= 0.0f) ? v : prelu_a[0] * v;
}

// ---------------------------------------------------------------------------
// Launch
// ---------------------------------------------------------------------------
extern "C" void kernel_launch(void* const* d_in, const int* in_sizes, int n_in,
                              void* d_out, int out_size, void* d_ws, size_t ws_size,
                              hipStream_t stream) {
    const float*     X  = (const float*)d_in[0];      // [N, 512]
    const long long* EI = (const long long*)d_in[1];  // [2, E] int64
    const float*     W  = (const float*)d_in[2];      // [512, 64]
    const float*     B  = (const float*)d_in[3];      // [64]
    const float*     A  = (const float*)d_in[4];      // [1]
    float* out = (float*)d_out;                       // [N, 64]

    // Workspace layout: dinv [N] floats, then x [N*64] floats (~26 MB total).
    float* dinv = (float*)d_ws;
    float* xbuf = dinv + N_NODES;

    // 0) zero accumulator (float4) + degree
    {
        int n4 = (N_NODES * OUT_DIM) / 4;   // 1.6M, also covers N_NODES scalar
        gcn_zero_kernel<<<(n4 + 255) / 256, 256, 0, stream>>>((float4*)out, dinv);
    }
    // 1) degree
    gcn_degree_kernel<<<(N_EDGES + 255) / 256, 256, 0, stream>>>(EI, dinv);
    // 2) d^{-1/2}
    gcn_dinv_kernel<<<(N_NODES + 255) / 256, 256, 0, stream>>>(dinv);
    // 3) x = X @ W  (WMMA f32): 6250 blocks x 128 threads
    gcn_gemm_kernel<<<N_NODES / 16, 128, 0, stream>>>(X, W, xbuf);
    // 4) scatter-add: 64 threads per edge (grid divides exactly: 800000 blocks)
    {
        long long total = (long long)N_EDGES * 64;
        int blocks = (int)((total + 255) / 256);
        gcn_scatter_kernel<<<blocks, 256, 0, stream>>>(EI, xbuf, dinv, out);
    }
    // 5) bias + PReLU
    {
        int n = N_NODES * OUT_DIM;
        gcn_bias_prelu_kernel<<<(n + 255) / 256, 256, 0, stream>>>(out, B, A);
    }
}